// MultiTaskFinancialGAT_19430432047546
// MI455X (gfx1250) — compile-verified
//
#include <hip/hip_runtime.h>
#include <hip/hip_bf16.h>

typedef __attribute__((ext_vector_type(2))) float v2f;
typedef __attribute__((ext_vector_type(8))) float v8f;

__device__ __forceinline__ void atomicMaxF(float* addr, float val) {
  int* ai = (int*)addr;
  int cur = __float_as_int(*addr);
  while (__int_as_float(cur) < val) {
    int assumed = cur;
    cur = atomicCAS(ai, assumed, __float_as_int(val));
    if (cur == assumed) break;
  }
}

__device__ __forceinline__ float warpSum(float v) {
#pragma unroll
  for (int m = 16; m >= 1; m >>= 1) v += __shfl_xor(v, m, 32);
  return v;
}

// ---------------- WMMA fp32 GEMM: C[nRows x 128] = A[nRows x K] @ B[K x 128]
// A tile (16 x K) staged in LDS once per block and shared by all 8 wave32s.
// mode 0: plain store; mode 1: input-projection epilogue (bias -> BN(eval) -> ReLU)
__global__ void k_gemm_wmma(const float* __restrict__ A, const float* __restrict__ B,
                            float* __restrict__ C, int nRows, int K,
                            const float* __restrict__ bias,
                            const float* __restrict__ bn_g,
                            const float* __restrict__ bn_b, int mode) {
  __shared__ float As[16 * 129];                 // stride 129 -> conflict-free column reads
  const int r0 = blockIdx.x * 16;
  for (int i = threadIdx.x; i < 16 * K; i += 256) {
    int r = i / K, c = i - r * K;
    int row = r0 + r;
    if (row >= nRows) row = nRows - 1;           // clamp (stores guarded below)
    As[r * 129 + c] = A[(size_t)row * K + c];
  }
  __syncthreads();

  const int lane = threadIdx.x & 31;
  const int wave = threadIdx.x >> 5;             // 8 waves -> 8 col tiles (128 cols)
  const int kb   = (lane >> 4) * 2;              // lanes 0-15: K 0,1 ; lanes 16-31: K 2,3
  const int n    = (wave << 4) + (lane & 15);
  const int mloc = lane & 15;
  v8f acc = {};
  for (int k = 0; k < K; k += 4) {
    v2f a, b;
    a.x = As[mloc * 129 + k + kb];
    a.y = As[mloc * 129 + k + kb + 1];
    b.x = B[(size_t)(k + kb) * 128 + n];
    b.y = B[(size_t)(k + kb + 1) * 128 + n];
    acc = __builtin_amdgcn_wmma_f32_16x16x4_f32(false, a, false, b, (short)0, acc,
                                                false, false);
  }
  const int rbase = r0 + ((lane >> 4) << 3);
#pragma unroll
  for (int v = 0; v < 8; ++v) {
    int row = rbase + v;
    if (row < nRows) {
      float val = acc[v];
      if (mode == 1) {
        float sc = bn_g[n] * rsqrtf(1.0f + 1e-5f);
        val = (val + bias[n]) * sc + bn_b[n];
        val = val > 0.f ? val : 0.f;
      }
      C[(size_t)row * 128 + n] = val;
    }
  }
}

// ---------------- graph preprocessing: degree + mean incoming edge_attr
__global__ void k_zero(float* p, long n) {
  long i = (long)blockIdx.x * blockDim.x + threadIdx.x;
  if (i < n) p[i] = 0.f;
}

__global__ void k_edge_degattr(const int* __restrict__ dst, const float* __restrict__ ea,
                               float* __restrict__ deg, float* __restrict__ lattr, int E) {
  int e = blockIdx.x * blockDim.x + threadIdx.x;
  if (e >= E) return;
  int d = dst[e];
  atomicAdd(deg + d, 1.0f);
#pragma unroll
  for (int j = 0; j < 10; ++j)
    atomicAdd(lattr + (size_t)d * 10 + j, ea[(size_t)e * 10 + j]);
}

__global__ void k_node_normattr(float* lattr, const float* deg, int N) {
  int n = blockIdx.x * blockDim.x + threadIdx.x;
  if (n >= N) return;
  float dg = deg[n];
  float inv = 1.0f / (dg > 1.0f ? dg : 1.0f);
#pragma unroll
  for (int j = 0; j < 10; ++j) lattr[(size_t)n * 10 + j] *= inv;
}

// ---------------- per-layer tiny matrix M[10,H] = sum_c We[d,h,c]*ae[h,c]
__global__ void k_compute_M(const float* __restrict__ We, const float* __restrict__ aew,
                            float* __restrict__ M, int H, int C) {
  int t = threadIdx.x;
  if (t >= 10 * H) return;
  int d = t / H, h = t % H;
  float s = 0.f;
  for (int c = 0; c < C; ++c) s += We[d * (H * C) + h * C + c] * aew[h * C + c];
  M[d * H + h] = s;
}

// ---------------- node: a_s, a_d, self-loop alpha; init amax
__global__ void k_node_attn_init(const float* __restrict__ xl,
                                 const float* __restrict__ asw, const float* __restrict__ adw,
                                 const float* __restrict__ lattr, const float* __restrict__ M,
                                 float* a_s, float* a_d, float* aloop, float* amax,
                                 int N, int H, int C) {
  int t = blockIdx.x * blockDim.x + threadIdx.x;
  if (t >= N * H) return;
  int n = t / H, h = t - n * H;
  const float* xr = xl + (size_t)n * 128 + h * C;
  float ss = 0.f, sd = 0.f;
  for (int c = 0; c < C; ++c) {
    float v = xr[c];
    ss += v * asw[h * C + c];
    sd += v * adw[h * C + c];
  }
  float se = 0.f;
#pragma unroll
  for (int j = 0; j < 10; ++j) se += lattr[(size_t)n * 10 + j] * M[j * H + h];
  a_s[t] = ss; a_d[t] = sd;
  float al = ss + sd + se;
  al = al > 0.f ? al : 0.2f * al;
  aloop[t] = al;
  amax[t]  = al;   // init (self-loop always present)
}

// ---------------- softmax pass A (fused a_e): alpha + atomicMax (eabuf <- alpha)
__global__ void k_edge_passA(const int* __restrict__ src, const int* __restrict__ dst,
                             const float* __restrict__ ea, const float* __restrict__ M,
                             const float* __restrict__ a_s, const float* __restrict__ a_d,
                             float* __restrict__ eabuf, float* __restrict__ amax,
                             int E, int H) {
  int e = blockIdx.x * blockDim.x + threadIdx.x;
  if (e >= E) return;
  float a[10];
#pragma unroll
  for (int j = 0; j < 10; ++j) a[j] = ea[(size_t)e * 10 + j];
  int s = src[e], d = dst[e];
  for (int h = 0; h < H; ++h) {
    float aeh = 0.f;
#pragma unroll
    for (int j = 0; j < 10; ++j) aeh += a[j] * M[j * H + h];
    float al = a_s[s * H + h] + a_d[d * H + h] + aeh;
    al = al > 0.f ? al : 0.2f * al;
    eabuf[(size_t)e * H + h] = al;
    atomicMaxF(amax + d * H + h, al);
  }
}

// ---------------- softmax pass B (node): self-loop exp; init denom
__global__ void k_node_passB(float* aloop, const float* amax, float* denom, int NH) {
  int t = blockIdx.x * blockDim.x + threadIdx.x;
  if (t >= NH) return;
  float ex = __expf(aloop[t] - amax[t]);
  aloop[t] = ex;
  denom[t] = ex;   // init
}

// ---------------- softmax pass B (edges): exp + atomicAdd denom (eabuf: alpha -> ex)
__global__ void k_edge_passB(const int* __restrict__ dst, float* __restrict__ eabuf,
                             const float* __restrict__ amax, float* __restrict__ denom,
                             int E, int H) {
  int e = blockIdx.x * blockDim.x + threadIdx.x;
  if (e >= E) return;
  int d = dst[e];
  for (int h = 0; h < H; ++h) {
    float ex = __expf(eabuf[(size_t)e * H + h] - amax[d * H + h]);
    eabuf[(size_t)e * H + h] = ex;
    atomicAdd(denom + d * H + h, ex);
  }
}

// ---------------- aggregation init (self-loop contribution; initializes out)
__global__ void k_node_outinit(const float* __restrict__ xl, const float* __restrict__ aloop,
                               const float* __restrict__ denom, float* __restrict__ out,
                               int N, int H, int log2C) {
  long t = (long)blockIdx.x * blockDim.x + threadIdx.x;
  if (t >= (long)N * 128) return;
  int n = (int)(t >> 7), ch = (int)(t & 127);
  int h = ch >> log2C;
  float coef = aloop[n * H + h] / (denom[n * H + h] + 1e-16f);
  out[t] = xl[t] * coef;
}

// ---------------- aggregation: wave per edge, scatter-add 128 channels
__global__ void k_edge_passC(const int* __restrict__ src, const int* __restrict__ dst,
                             const float* __restrict__ xl, const float* __restrict__ eabuf,
                             const float* __restrict__ denom, float* __restrict__ out,
                             int E, int H, int log2C) {
  int lane = threadIdx.x & 31;
  int e = blockIdx.x * 8 + (threadIdx.x >> 5);
  if (e >= E) return;
  int s = src[e], d = dst[e];
#pragma unroll
  for (int i = 0; i < 4; ++i) {
    int ch = lane + i * 32;
    int h  = ch >> log2C;
    float coef = eabuf[(size_t)e * H + h] / (denom[d * H + h] + 1e-16f);
    atomicAdd(out + (size_t)d * 128 + ch, xl[(size_t)s * 128 + ch] * coef);
  }
}

// ---------------- bias -> ReLU -> residual -> LayerNorm (wave per node)
__global__ void k_node_post(const float* __restrict__ out, const float* __restrict__ bias,
                            const float* __restrict__ hin, const float* __restrict__ g,
                            const float* __restrict__ b, float* __restrict__ hout, int N) {
  int lane = threadIdx.x & 31;
  int n = blockIdx.x * 8 + (threadIdx.x >> 5);
  if (n >= N) return;
  float v[4];
  float s = 0.f;
#pragma unroll
  for (int i = 0; i < 4; ++i) {
    int ch = lane + i * 32;
    float t = out[(size_t)n * 128 + ch] + bias[ch];
    t = t > 0.f ? t : 0.f;
    t += hin[(size_t)n * 128 + ch];
    v[i] = t; s += t;
  }
  s = warpSum(s);
  float mu = s * (1.0f / 128.0f);
  float q = 0.f;
#pragma unroll
  for (int i = 0; i < 4; ++i) { float dv = v[i] - mu; q += dv * dv; }
  q = warpSum(q);
  float rinv = rsqrtf(q * (1.0f / 128.0f) + 1e-5f);
#pragma unroll
  for (int i = 0; i < 4; ++i) {
    int ch = lane + i * 32;
    hout[(size_t)n * 128 + ch] = (v[i] - mu) * rinv * g[ch] + b[ch];
  }
}

// ---------------- default-probability head (wave per node)
__global__ void k_dp_head(const float* __restrict__ h, const float* __restrict__ w1,
                          const float* __restrict__ b1, const float* __restrict__ w2,
                          const float* __restrict__ b2, float* __restrict__ logits, int N) {
  int lane = threadIdx.x & 31;
  int n = blockIdx.x * 8 + (threadIdx.x >> 5);
  if (n >= N) return;
  float total = 0.f;
#pragma unroll
  for (int jj = 0; jj < 2; ++jj) {
    int j = lane + jj * 32;
    float acc = b1[j];
    for (int k = 0; k < 128; ++k) acc += h[(size_t)n * 128 + k] * w1[k * 64 + j];
    acc = acc > 0.f ? acc : 0.f;
    total += acc * w2[j];
  }
  total = warpSum(total);
  if (lane == 0) logits[n] = total + b2[0];
}

// ---------------- global pooling (sum + max per channel)
__global__ void k_pool_init(float* pool) {
  int t = threadIdx.x;
  if (t < 128) { pool[t] = 0.f; pool[128 + t] = -3.4e38f; }
}

__global__ void k_pool(const float* __restrict__ h, float* __restrict__ pool, int N, int rpb) {
  int t = threadIdx.x;           // 256
  int ch = t & 127;
  int half = t >> 7;
  int n0 = blockIdx.x * rpb + half;
  int n1 = blockIdx.x * rpb + rpb; if (n1 > N) n1 = N;
  float s = 0.f, mx = -3.4e38f;
  for (int n = n0; n < n1; n += 2) {
    float v = h[(size_t)n * 128 + ch];
    s += v; mx = mx > v ? mx : v;
  }
  atomicAdd(pool + ch, s);
  atomicMaxF(pool + 128 + ch, mx);
}

// ---------------- systemic-risk head (single block)
__global__ void k_systemic(const float* __restrict__ pool, const float* __restrict__ w1,
                           const float* __restrict__ b1, const float* __restrict__ w2,
                           const float* __restrict__ b2, float* __restrict__ outp, int N) {
  __shared__ float red[128];
  int j = threadIdx.x;           // 128
  float acc = b1[j];
  float invN = 1.0f / (float)N;
  for (int k = 0; k < 128; ++k) acc += (pool[k] * invN) * w1[k * 128 + j];          // mean
  for (int k = 0; k < 128; ++k) acc += pool[128 + k] * w1[(128 + k) * 128 + j];     // max
  for (int k = 0; k < 128; ++k) acc += pool[k] * w1[(256 + k) * 128 + j];           // sum
  acc = acc > 0.f ? acc : 0.f;
  red[j] = acc * w2[j];
  __syncthreads();
  for (int st = 64; st >= 1; st >>= 1) {
    if (j < st) red[j] += red[j + st];
    __syncthreads();
  }
  if (j == 0) outp[0] = red[0] + b2[0];
}

extern "C" void kernel_launch(void* const* d_in, const int* in_sizes, int n_in,
                              void* d_out, int out_size, void* d_ws, size_t ws_size,
                              hipStream_t stream) {
  const float* x    = (const float*)d_in[0];
  const int*   ei   = (const int*)d_in[1];
  const float* ea   = (const float*)d_in[2];
  const float* w_in = (const float*)d_in[3];
  const float* b_in = (const float*)d_in[4];
  const float* bn_g = (const float*)d_in[5];
  const float* bn_b = (const float*)d_in[6];
  const float* dp_w1 = (const float*)d_in[39];
  const float* dp_b1 = (const float*)d_in[40];
  const float* dp_w2 = (const float*)d_in[41];
  const float* dp_b2 = (const float*)d_in[42];
  const float* sr_w1 = (const float*)d_in[43];
  const float* sr_b1 = (const float*)d_in[44];
  const float* sr_w2 = (const float*)d_in[45];
  const float* sr_b2 = (const float*)d_in[46];

  const int N = in_sizes[0] / 64;
  const int E = in_sizes[1] / 2;
  const int* src = ei;
  const int* dst = ei + E;

  // workspace layout
  float* ws = (float*)d_ws;
  size_t off = 0;
  float* h     = ws + off; off += (size_t)N * 128;
  float* xl    = ws + off; off += (size_t)N * 128;
  float* outb  = ws + off; off += (size_t)N * 128;
  float* a_s   = ws + off; off += (size_t)N * 8;
  float* a_d   = ws + off; off += (size_t)N * 8;
  float* lattr = ws + off; off += (size_t)N * 10;   // lattr+deg contiguous (zeroed together)
  float* deg   = ws + off; off += (size_t)N;
  float* aloop = ws + off; off += (size_t)N * 8;
  float* amax  = ws + off; off += (size_t)N * 8;
  float* denom = ws + off; off += (size_t)N * 8;
  float* eabuf = ws + off; off += (size_t)E * 8;    // alpha -> ex (in place)
  float* M     = ws + off; off += 128;
  float* pool  = ws + off; off += 256;

  float* d_logits = (float*)d_out;
  float* d_sys    = d_logits + N;
  float* d_h      = d_sys + 1;

  // --- preprocessing: deg + mean incoming edge attr (self-loop attrs)
  long zN = (long)N * 11;
  k_zero<<<(int)((zN + 255) / 256), 256, 0, stream>>>(lattr, zN);
  k_edge_degattr<<<(E + 255) / 256, 256, 0, stream>>>(dst, ea, deg, lattr, E);
  k_node_normattr<<<(N + 255) / 256, 256, 0, stream>>>(lattr, deg, N);

  // --- input projection: Linear -> BN(eval) -> ReLU (WMMA fp32)
  k_gemm_wmma<<<(N + 15) / 16, 256, 0, stream>>>(x, w_in, h, N, 64, b_in, bn_g, bn_b, 1);

  const int Hs[4]  = {8, 8, 8, 1};
  const int Cs[4]  = {16, 16, 16, 128};
  const int L2C[4] = {4, 4, 4, 7};
  for (int l = 0; l < 4; ++l) {
    const float* W    = (const float*)d_in[7 + 8 * l + 0];
    const float* asw  = (const float*)d_in[7 + 8 * l + 1];
    const float* adw  = (const float*)d_in[7 + 8 * l + 2];
    const float* Wew  = (const float*)d_in[7 + 8 * l + 3];
    const float* aew  = (const float*)d_in[7 + 8 * l + 4];
    const float* bias = (const float*)d_in[7 + 8 * l + 5];
    const float* lng  = (const float*)d_in[7 + 8 * l + 6];
    const float* lnb  = (const float*)d_in[7 + 8 * l + 7];
    int H = Hs[l], C = Cs[l], lc = L2C[l];

    k_compute_M<<<1, 128, 0, stream>>>(Wew, aew, M, H, C);
    k_gemm_wmma<<<(N + 15) / 16, 256, 0, stream>>>(h, W, xl, N, 128,
                                                   nullptr, nullptr, nullptr, 0);
    k_node_attn_init<<<(N * H + 255) / 256, 256, 0, stream>>>(xl, asw, adw, lattr, M,
                                                              a_s, a_d, aloop, amax, N, H, C);
    k_edge_passA<<<(E + 255) / 256, 256, 0, stream>>>(src, dst, ea, M, a_s, a_d,
                                                      eabuf, amax, E, H);
    k_node_passB<<<(N * H + 255) / 256, 256, 0, stream>>>(aloop, amax, denom, N * H);
    k_edge_passB<<<(E + 255) / 256, 256, 0, stream>>>(dst, eabuf, amax, denom, E, H);
    long t128 = (long)N * 128;
    k_node_outinit<<<(int)((t128 + 255) / 256), 256, 0, stream>>>(xl, aloop, denom, outb,
                                                                  N, H, lc);
    k_edge_passC<<<(E + 7) / 8, 256, 0, stream>>>(src, dst, xl, eabuf, denom, outb, E, H, lc);
    float* hout = (l == 3) ? d_h : h;
    k_node_post<<<(N + 7) / 8, 256, 0, stream>>>(outb, bias, h, lng, lnb, hout, N);
  }

  // --- heads
  k_dp_head<<<(N + 7) / 8, 256, 0, stream>>>(d_h, dp_w1, dp_b1, dp_w2, dp_b2, d_logits, N);
  k_pool_init<<<1, 128, 0, stream>>>(pool);
  int rpb = (N + 511) / 512;
  k_pool<<<512, 256, 0, stream>>>(d_h, pool, N, rpb);
  k_systemic<<<1, 128, 0, stream>>>(pool, sr_w1, sr_b1, sr_w2, sr_b2, d_sys, N);
}